// GraphSAGEEncoder_67413806678196
// MI455X (gfx1250) — compile-verified
//
#include <hip/hip_runtime.h>
#include <hip/hip_bf16.h>

typedef __attribute__((ext_vector_type(2))) float v2f;
typedef __attribute__((ext_vector_type(8))) float v8f;

// Reduce across the 16 lanes of each half of a wave32 (rows of a WMMA 16x16
// f32 C/D tile live entirely within one half-wave).
__device__ __forceinline__ float half16_sum(float v) {
  v += __shfl_xor(v, 1, 16);
  v += __shfl_xor(v, 2, 16);
  v += __shfl_xor(v, 4, 16);
  v += __shfl_xor(v, 8, 16);
  return v;
}

__global__ void sage_count_kernel(const long long* __restrict__ ei,
                                  float* __restrict__ cnt, long long E) {
  const long long stride = (long long)gridDim.x * blockDim.x;
  for (long long e = (long long)blockIdx.x * blockDim.x + threadIdx.x; e < E; e += stride) {
    atomicAdd(&cnt[(int)ei[E + e]], 1.0f);
  }
}

template <int F>
__global__ void sage_scatter_kernel(const float* __restrict__ h,
                                    const long long* __restrict__ ei,
                                    float* __restrict__ agg, long long E) {
  const long long total = E * (long long)F;
  const long long stride = (long long)gridDim.x * blockDim.x;
  for (long long i = (long long)blockIdx.x * blockDim.x + threadIdx.x; i < total; i += stride) {
    const long long e = i / F;
    const int f = (int)(i - e * (long long)F);
    const int s = (int)ei[e];
    const int d = (int)ei[E + e];
    atomicAdd(&agg[(size_t)d * F + f], h[(size_t)s * F + f]);
  }
}

// One wave32 per 16-node tile. D = mean@Wl + b + x@Wr via V_WMMA_F32_16X16X4_F32,
// then L2 row-normalize, optionally LayerNorm+ReLU, store.
template <int F_IN, int F_OUT, bool LN_RELU>
__global__ __launch_bounds__(256) void sage_node_kernel(
    const float* __restrict__ x, const float* __restrict__ agg,
    const float* __restrict__ cnt,
    const float* __restrict__ Wl, const float* __restrict__ Wr,
    const float* __restrict__ bias, const float* __restrict__ g,
    const float* __restrict__ be, float* __restrict__ out, int N) {
  constexpr int NT = F_OUT / 16;  // 16-col output tiles
  constexpr int KC = F_IN / 4;    // K chunks of 4

  __shared__ float sWl[F_IN * F_OUT];
  __shared__ float sWr[F_IN * F_OUT];
  __shared__ float sB[F_OUT];
  __shared__ float sG[F_OUT];
  __shared__ float sBe[F_OUT];
  for (int i = threadIdx.x; i < F_IN * F_OUT; i += blockDim.x) {
    sWl[i] = Wl[i];
    sWr[i] = Wr[i];
  }
  for (int i = threadIdx.x; i < F_OUT; i += blockDim.x) {
    sB[i] = bias[i];
    if constexpr (LN_RELU) { sG[i] = g[i]; sBe[i] = be[i]; }
    else                   { sG[i] = 1.f;  sBe[i] = 0.f;   }
  }
  __syncthreads();

  const int lane = threadIdx.x & 31;
  const int wave = threadIdx.x >> 5;
  const int tile = blockIdx.x * (blockDim.x >> 5) + wave;
  const int base = tile * 16;
  if (base >= N) return;

  const int rlo = lane & 15;  // row (for A) / col-in-tile (for B,C,D)
  const int hi = lane >> 4;   // half-wave select
  int rowA = base + rlo;
  if (rowA > N - 1) rowA = N - 1;
  const float rc = 1.0f / fmaxf(cnt[rowA], 1.0f);  // 1/deg (clamped)
  const float* aggRow = agg + (size_t)rowA * F_IN;
  const float* xRow = x + (size_t)rowA * F_IN;

  v8f acc[NT] = {};

  // mean @ Wl
#pragma unroll
  for (int c = 0; c < KC; ++c) {
    const int k0 = 4 * c + 2 * hi;  // A layout: lanes 0-15 K={0,1}, lanes 16-31 K={2,3}
    v2f a;
    a.x = aggRow[k0] * rc;
    a.y = aggRow[k0 + 1] * rc;
#pragma unroll
    for (int t = 0; t < NT; ++t) {
      const int col = t * 16 + rlo;
      v2f bf;
      bf.x = sWl[k0 * F_OUT + col];
      bf.y = sWl[(k0 + 1) * F_OUT + col];
      acc[t] = __builtin_amdgcn_wmma_f32_16x16x4_f32(false, a, false, bf, (short)0,
                                                     acc[t], false, false);
    }
  }
  // x @ Wr
#pragma unroll
  for (int c = 0; c < KC; ++c) {
    const int k0 = 4 * c + 2 * hi;
    v2f a;
    a.x = xRow[k0];
    a.y = xRow[k0 + 1];
#pragma unroll
    for (int t = 0; t < NT; ++t) {
      const int col = t * 16 + rlo;
      v2f bf;
      bf.x = sWr[k0 * F_OUT + col];
      bf.y = sWr[(k0 + 1) * F_OUT + col];
      acc[t] = __builtin_amdgcn_wmma_f32_16x16x4_f32(false, a, false, bf, (short)0,
                                                     acc[t], false, false);
    }
  }

  // + bias (depends on column only)
#pragma unroll
  for (int t = 0; t < NT; ++t) {
    const float bc = sB[t * 16 + rlo];
#pragma unroll
    for (int j = 0; j < 8; ++j) acc[t][j] += bc;
  }

  // Row-wise L2 normalization (row j+8*hi lives in this half-wave at VGPR j)
#pragma unroll
  for (int j = 0; j < 8; ++j) {
    float s = 0.f;
#pragma unroll
    for (int t = 0; t < NT; ++t) s += acc[t][j] * acc[t][j];
    s = half16_sum(s);
    const float rn = 1.0f / fmaxf(sqrtf(s), 1e-12f);
#pragma unroll
    for (int t = 0; t < NT; ++t) acc[t][j] *= rn;
  }

  if constexpr (LN_RELU) {
#pragma unroll
    for (int j = 0; j < 8; ++j) {
      float s1 = 0.f, s2 = 0.f;
#pragma unroll
      for (int t = 0; t < NT; ++t) {
        const float v = acc[t][j];
        s1 += v;
        s2 += v * v;
      }
      s1 = half16_sum(s1);
      s2 = half16_sum(s2);
      const float mu = s1 / (float)F_OUT;
      const float var = s2 / (float)F_OUT - mu * mu;
      const float rs = rsqrtf(var + 1e-5f);
#pragma unroll
      for (int t = 0; t < NT; ++t) {
        const int col = t * 16 + rlo;
        const float v = (acc[t][j] - mu) * rs * sG[col] + sBe[col];
        acc[t][j] = fmaxf(v, 0.f);
      }
    }
  }

  // Store: VGPR j holds row (base + j + 8*hi), cols t*16 + rlo
#pragma unroll
  for (int j = 0; j < 8; ++j) {
    const int row = base + j + hi * 8;
    if (row < N) {
      float* o = out + (size_t)row * F_OUT;
#pragma unroll
      for (int t = 0; t < NT; ++t) o[t * 16 + rlo] = acc[t][j];
    }
  }
}

extern "C" void kernel_launch(void* const* d_in, const int* in_sizes, int n_in,
                              void* d_out, int out_size, void* d_ws, size_t ws_size,
                              hipStream_t stream) {
  const float* x = (const float*)d_in[0];
  const long long* ei = (const long long*)d_in[1];  // int64 [2, E]
  const float* Wl1 = (const float*)d_in[2];
  const float* Wr1 = (const float*)d_in[3];
  const float* b1 = (const float*)d_in[4];
  const float* g1 = (const float*)d_in[5];
  const float* be1 = (const float*)d_in[6];
  const float* Wl2 = (const float*)d_in[7];
  const float* Wr2 = (const float*)d_in[8];
  const float* b2 = (const float*)d_in[9];
  const float* g2 = (const float*)d_in[10];
  const float* be2 = (const float*)d_in[11];
  const float* Wl3 = (const float*)d_in[12];
  const float* Wr3 = (const float*)d_in[13];
  const float* b3 = (const float*)d_in[14];

  const int IN = 12, H = 64;
  const int N = in_sizes[0] / IN;
  const long long E = (long long)in_sizes[1] / 2;

  float* cnt = (float*)d_ws;                 // N
  float* agg = cnt + N;                      // N*64 (reused each layer)
  float* h1 = agg + (size_t)N * H;           // N*64
  float* h2 = h1 + (size_t)N * H;            // N*64

  const int tiles = (N + 15) / 16;
  const int nodeBlocks = (tiles + 7) / 8;  // 8 waves (tiles) per 256-thread block

  // degree counts
  hipMemsetAsync(cnt, 0, (size_t)N * sizeof(float), stream);
  sage_count_kernel<<<2048, 256, 0, stream>>>(ei, cnt, E);

  // layer 1: 12 -> 64, L2 norm + LN + ReLU
  hipMemsetAsync(agg, 0, (size_t)N * IN * sizeof(float), stream);
  sage_scatter_kernel<12><<<8192, 256, 0, stream>>>(x, ei, agg, E);
  sage_node_kernel<12, 64, true>
      <<<nodeBlocks, 256, 0, stream>>>(x, agg, cnt, Wl1, Wr1, b1, g1, be1, h1, N);

  // layer 2: 64 -> 64, L2 norm + LN + ReLU
  hipMemsetAsync(agg, 0, (size_t)N * H * sizeof(float), stream);
  sage_scatter_kernel<64><<<16384, 256, 0, stream>>>(h1, ei, agg, E);
  sage_node_kernel<64, 64, true>
      <<<nodeBlocks, 256, 0, stream>>>(h1, agg, cnt, Wl2, Wr2, b2, g2, be2, h2, N);

  // layer 3: 64 -> 32, L2 norm only, straight to d_out
  hipMemsetAsync(agg, 0, (size_t)N * H * sizeof(float), stream);
  sage_scatter_kernel<64><<<16384, 256, 0, stream>>>(h2, ei, agg, E);
  sage_node_kernel<64, 32, false>
      <<<nodeBlocks, 256, 0, stream>>>(h2, agg, cnt, Wl3, Wr3, b3, nullptr, nullptr,
                                       (float*)d_out, N);
}